// QuBlur_712964571922
// MI455X (gfx1250) — compile-verified
//
#include <hip/hip_runtime.h>
#include <math.h>

typedef __attribute__((ext_vector_type(2))) float v2f;
typedef __attribute__((ext_vector_type(8))) float v8f;

#define NTOT 4096            // matrix edge
#define WAVES_PER_WG 8       // 256 threads, wave32
#define WG_COUNT 8192        // 1024 block-rows * 64 tile-cols / 8 waves
// d_ws float layout
#define WS_MRE   0           // 256 floats: M_re (16x16 row-major)
#define WS_MIM   256         // 256 floats: M_im
#define WS_TOTAL 512         // 1 float: global sum of squares
#define WS_PART  1024        // WG_COUNT floats: per-workgroup partials

// ---------------------------------------------------------------------------
// Kernel 0: build M = U (x) conj(U) as two real 16x16 matrices in workspace.
// U = G3 @ RYY(w2) @ G1 = c2*(G3@G1) - i*s2*(G3@YY@G1), all-real factors.
// ---------------------------------------------------------------------------
__global__ __launch_bounds__(256) void qblur_setup(const float* __restrict__ weight,
                                                   float* __restrict__ wsf) {
    const float WMUL = 0.6324555320336759f;   // sqrt(2)*5^-0.5
    float w[5];
    #pragma unroll
    for (int k = 0; k < 5; ++k) w[k] = weight[k] * WMUL * 0.5f;  // half-angles
    float c0 = cosf(w[0]), s0 = sinf(w[0]);
    float c1 = cosf(w[1]), s1 = sinf(w[1]);
    float c2 = cosf(w[2]), s2 = sinf(w[2]);
    float c3 = cosf(w[3]), s3 = sinf(w[3]);
    float c4 = cosf(w[4]), s4 = sinf(w[4]);
    float R0[2][2] = {{c0, -s0}, {s0, c0}};
    float R1[2][2] = {{c1, -s1}, {s1, c1}};
    float R3[2][2] = {{c3, -s3}, {s3, c3}};
    float R4[2][2] = {{c4, -s4}, {s4, c4}};
    float G1[4][4], G3[4][4];
    #pragma unroll
    for (int p = 0; p < 2; ++p)
      #pragma unroll
      for (int q = 0; q < 2; ++q)
        #pragma unroll
        for (int r = 0; r < 2; ++r)
          #pragma unroll
          for (int s = 0; s < 2; ++s) {
            G1[p*2+q][r*2+s] = R0[p][r] * R1[q][s];
            G3[p*2+q][r*2+s] = R3[p][r] * R4[q][s];
          }
    const float YY[4][4] = {{0,0,0,-1},{0,0,1,0},{0,1,0,0},{-1,0,0,0}};
    float P[4][4], T[4][4], Q[4][4], Ure[4][4], Uim[4][4];
    #pragma unroll
    for (int i = 0; i < 4; ++i)
      #pragma unroll
      for (int j = 0; j < 4; ++j) {
        float sp = 0.f, st = 0.f;
        #pragma unroll
        for (int k = 0; k < 4; ++k) { sp += G3[i][k]*G1[k][j]; st += YY[i][k]*G1[k][j]; }
        P[i][j] = sp; T[i][j] = st;
      }
    #pragma unroll
    for (int i = 0; i < 4; ++i)
      #pragma unroll
      for (int j = 0; j < 4; ++j) {
        float sq = 0.f;
        #pragma unroll
        for (int k = 0; k < 4; ++k) sq += G3[i][k]*T[k][j];
        Q[i][j] = sq;
        Ure[i][j] = c2 * P[i][j];
        Uim[i][j] = -s2 * sq;
      }
    // thread t -> M[r][c], r = out component (i,j), c = in component (k,l)
    int t = threadIdx.x;
    int r = t >> 4, c = t & 15;
    int i = r >> 2, j = r & 3, k = c >> 2, l = c & 3;
    wsf[WS_MRE + t] = Ure[i][k]*Ure[j][l] + Uim[i][k]*Uim[j][l];
    wsf[WS_MIM + t] = Uim[i][k]*Ure[j][l] - Ure[i][k]*Uim[j][l];
}

// ---------------------------------------------------------------------------
// Kernel 1: per wave, 16 blocks (4 rows x 64 cols of x).
//   B operand (4x16, f32, 2 VGPRs): lane L (hi=L>>4, n=L&15) holds block n,
//   rows i=q (chunk), cols j = 2*hi + v.  A operand = 16x4 panel of M.
// ---------------------------------------------------------------------------
__global__ __launch_bounds__(256) void qblur_main(const float* __restrict__ x,
                                                  const float* __restrict__ wsf,
                                                  float* __restrict__ out,
                                                  float* __restrict__ partials) {
    __shared__ float lds[WAVES_PER_WG][2][256];
    __shared__ float wsum[WAVES_PER_WG];

    const int tid  = threadIdx.x;
    const int wv   = tid >> 5;
    const int lane = tid & 31;
    const int hi   = lane >> 4;
    const int l16  = lane & 15;

    const unsigned gw = blockIdx.x * WAVES_PER_WG + wv;   // global wave index
    const unsigned a  = gw >> 6;                          // block-row (0..1023)
    const unsigned tc = gw & 63;                          // tile-col (0..63)

    const size_t rowbase = (size_t)(4u * a) * NTOT;
    const unsigned col   = tc * 64u + (unsigned)(l16 * 4 + hi * 2);

    // ---- load + block norms ----
    v2f b[4];
    float nsq = 0.f;
    #pragma unroll
    for (int q = 0; q < 4; ++q) {
        const float* p = x + rowbase + (size_t)q * NTOT + col;
        float e0 = p[0], e1 = p[1];
        b[q][0] = e0; b[q][1] = e1;
        nsq += e0*e0 + e1*e1;
    }
    nsq += __shfl_xor(nsq, 16);                 // combine the two half-wave lanes
    const float inv = (nsq > 0.f) ? (1.0f / sqrtf(nsq)) : 0.f;
    #pragma unroll
    for (int q = 0; q < 4; ++q) { b[q][0] *= inv; b[q][1] *= inv; }

    // ---- A operands: 16x4 panels of M_re / M_im ----
    const int abase = l16 * 16 + hi * 2;
    v2f are[4], aim[4];
    #pragma unroll
    for (int q = 0; q < 4; ++q) {
        are[q][0] = wsf[WS_MRE + abase + 4*q];
        are[q][1] = wsf[WS_MRE + abase + 4*q + 1];
        aim[q][0] = wsf[WS_MIM + abase + 4*q];
        aim[q][1] = wsf[WS_MIM + abase + 4*q + 1];
    }

    // ---- 8 chained f32 WMMAs: D = sum_q A_q * B_q ----
    v8f dre = {};
    v8f dim = {};
    #pragma unroll
    for (int q = 0; q < 4; ++q) {
        dre = __builtin_amdgcn_wmma_f32_16x16x4_f32(false, are[q], false, b[q],
                                                    (short)0, dre, false, false);
        dim = __builtin_amdgcn_wmma_f32_16x16x4_f32(false, aim[q], false, b[q],
                                                    (short)0, dim, false, false);
    }

    // ---- partial sum of squares (for global Frobenius norm) ----
    float s = 0.f;
    #pragma unroll
    for (int p = 0; p < 8; ++p) s += dre[p]*dre[p] + dim[p]*dim[p];
    #pragma unroll
    for (int off = 16; off; off >>= 1) s += __shfl_xor(s, off);
    if (lane == 0) wsum[wv] = s;

    // ---- stage D through LDS: lane L, vgpr p holds out row (8*hi+p), col l16 ----
    float* lre = &lds[wv][0][0];
    float* lim = &lds[wv][1][0];
    const int st0 = (2*hi) * 64 + 4*l16;        // rows 2*hi and 2*hi+1, 4 floats each
    *(float4*)(lre + st0)      = make_float4(dre[0], dre[1], dre[2], dre[3]);
    *(float4*)(lre + st0 + 64) = make_float4(dre[4], dre[5], dre[6], dre[7]);
    *(float4*)(lim + st0)      = make_float4(dim[0], dim[1], dim[2], dim[3]);
    *(float4*)(lim + st0 + 64) = make_float4(dim[4], dim[5], dim[6], dim[7]);
    __syncthreads();

    if (tid == 0) {
        float tot = 0.f;
        #pragma unroll
        for (int w = 0; w < WAVES_PER_WG; ++w) tot += wsum[w];
        partials[blockIdx.x] = tot;
    }

    // ---- contiguous 256B-per-row global stores ----
    const int f0 = 4 * lane;                    // flat float index in 4x64 tile
    const int f1 = f0 + 128;
    const size_t o0 = (size_t)(4u*a + (unsigned)(f0 >> 6)) * NTOT + tc*64u + (unsigned)(f0 & 63);
    const size_t o1 = (size_t)(4u*a + (unsigned)(f1 >> 6)) * NTOT + tc*64u + (unsigned)(f1 & 63);
    float* outim = out + (size_t)NTOT * NTOT;
    *(float4*)(out   + o0) = *(const float4*)(lre + f0);
    *(float4*)(out   + o1) = *(const float4*)(lre + f1);
    *(float4*)(outim + o0) = *(const float4*)(lim + f0);
    *(float4*)(outim + o1) = *(const float4*)(lim + f1);
}

// ---------------------------------------------------------------------------
// Kernel 2: deterministic fixed-order reduction of per-WG partials.
// ---------------------------------------------------------------------------
__global__ __launch_bounds__(256) void qblur_reduce(float* __restrict__ wsf) {
    __shared__ float sm[256];
    const int t = threadIdx.x;
    float s = 0.f;
    for (int i = t; i < WG_COUNT; i += 256) s += wsf[WS_PART + i];
    sm[t] = s;
    __syncthreads();
    for (int off = 128; off; off >>= 1) {
        if (t < off) sm[t] += sm[t + off];
        __syncthreads();
    }
    if (t == 0) wsf[WS_TOTAL] = sm[0];
}

// ---------------------------------------------------------------------------
// Kernel 3: scale both planes by 1/sqrt(total).
// ---------------------------------------------------------------------------
__global__ __launch_bounds__(256) void qblur_scale(float* __restrict__ out,
                                                   const float* __restrict__ wsf) {
    const float inv = 1.0f / sqrtf(wsf[WS_TOTAL]);
    const size_t idx = ((size_t)blockIdx.x * 256u + threadIdx.x) * 4u;
    float4 v = *(float4*)(out + idx);
    v.x *= inv; v.y *= inv; v.z *= inv; v.w *= inv;
    *(float4*)(out + idx) = v;
}

extern "C" void kernel_launch(void* const* d_in, const int* in_sizes, int n_in,
                              void* d_out, int out_size, void* d_ws, size_t ws_size,
                              hipStream_t stream) {
    const float* x      = (const float*)d_in[0];   // [4096,4096] f32
    const float* weight = (const float*)d_in[1];   // [5] f32
    float* out = (float*)d_out;                    // [2,4096,4096] f32
    float* wsf = (float*)d_ws;

    qblur_setup<<<1, 256, 0, stream>>>(weight, wsf);
    qblur_main<<<WG_COUNT, 256, 0, stream>>>(x, wsf, out, wsf + WS_PART);
    qblur_reduce<<<1, 256, 0, stream>>>(wsf);
    // 2*4096*4096 floats / 4 per thread / 256 threads = 32768 blocks
    qblur_scale<<<32768, 256, 0, stream>>>(out, wsf);
}